// ConvOffset_26439818674367
// MI455X (gfx1250) — compile-verified
//
#include <hip/hip_runtime.h>

typedef __attribute__((ext_vector_type(2))) float v2f;
typedef __attribute__((ext_vector_type(4))) float v4f;
typedef __attribute__((ext_vector_type(8))) float v8f;
typedef __attribute__((ext_vector_type(4))) int   v4i;

#define CIN  128
#define COUT 128
#define Bn   8
#define Hn   256
#define Wn   256
#define TAPS 9

#define AS1 __attribute__((address_space(1)))
#define AS3 __attribute__((address_space(3)))

// ---------------------------------------------------------------------------
// Fold conv kernel (3x3x128x128, HWIO) with dense W (128x128) into per-tap
// A_t = K_t @ W stored PRE-SWIZZLED for the conv kernel's LDS layout:
//   As[t][(k>>2)*512 + n*4 + (k&3)] = A_t[k][n]
// Also: bias2 = bias @ W, and exact per-tap nonzero flags (data-driven
// tap skipping; for the one-hot conv kernel only 1 of 9 taps survives,
// which is what keeps this op on the 23.3 TB/s memory roofline).
// ---------------------------------------------------------------------------
__global__ __launch_bounds__(256) void fold_kernel(
    const float* __restrict__ kern,   // (3,3,128,128)
    const float* __restrict__ bias,   // (128)
    const float* __restrict__ Wd,     // (128,128)
    float* __restrict__ As,           // (9,128,128) swizzled
    float* __restrict__ bias2,        // (128)
    int*   __restrict__ flags)        // (9)
{
    const int t   = blockIdx.x;       // tap 0..8
    const int tid = threadIdx.x;      // 0..255
    const float* Kt = kern + (size_t)t * CIN * COUT;

    // exact nonzero detection on the raw conv kernel tap
    __shared__ int s_any[256];
    int any = 0;
    for (int i = tid; i < CIN * COUT; i += 256) any |= (Kt[i] != 0.0f);
    s_any[tid] = any;
    __syncthreads();
    for (int s = 128; s > 0; s >>= 1) {
        if (tid < s) s_any[tid] |= s_any[tid + s];
        __syncthreads();
    }
    if (tid == 0) flags[t] = s_any[0];

    // A_t = Kt (cin x c) @ Wd (c x o), written swizzled
    for (int idx = tid; idx < CIN * COUT; idx += 256) {
        const int ci = idx >> 7;
        const int o  = idx & 127;
        float acc = 0.0f;
        for (int c = 0; c < COUT; ++c)
            acc += Kt[ci * COUT + c] * Wd[c * COUT + o];
        As[(size_t)t * CIN * COUT + (ci >> 2) * 512 + o * 4 + (ci & 3)] = acc;
    }

    if (t == 0) {
        for (int o = tid; o < COUT; o += 256) {
            float acc = 0.0f;
            for (int c = 0; c < COUT; ++c) acc += bias[c] * Wd[c * COUT + o];
            bias2[o] = acc;
        }
    }
}

// ---------------------------------------------------------------------------
// Implicit-GEMM conv with folded weights, f32 WMMA 16x16x4 (exact fp32).
// Workgroup = 8 waves = 128 consecutive pixels (half an image row).
// Wave = 16 pixels (M) x 128 couts (N) = 8 WMMA tiles, K = 128 per tap.
// A_t arrives pre-swizzled, so staging is a contiguous 64 KB copy done
// with GLOBAL_LOAD_ASYNC_TO_LDS_B128 (ASYNCcnt) when available.
// B-fragment reads are single ds_load_b64 per tile, bank-conflict-free
// (lane (h,n) touches banks {4n+2h, 4n+2h+1} -> all 64 banks exactly once).
// ---------------------------------------------------------------------------
__global__ __launch_bounds__(256) void conv_wmma_kernel(
    const float* __restrict__ x,      // (8,256,256,128)
    const float* __restrict__ As,     // (9,128,128) swizzled
    const float* __restrict__ bias2,  // (128)
    const int*   __restrict__ flags,  // (9)
    float* __restrict__ out)          // (8,256,256,128)
{
    __shared__ float sB[CIN * COUT];  // 65536 B, swizzled [k>>2][n][k&3]

    const int tid  = threadIdx.x;
    const int wave = tid >> 5;        // 0..7
    const int lane = tid & 31;
    const int lm   = lane & 15;       // M-row (A) / N-col (B,C,D) within half
    const int lh   = lane >> 4;       // half-wave: selects K pair / M+8 rows

    const long pixBase = (long)blockIdx.x * 128;
    const int  w0g = (int)(pixBase % Wn);             // 0 or 128
    const int  h0  = (int)((pixBase / Wn) % Hn);
    const int  b0  = (int)(pixBase / ((long)Wn * Hn));
    const int  wt  = w0g + wave * 16;                 // wave's first w

    // init accumulators with folded bias (same value for every M row)
    v8f acc[8];
    #pragma unroll
    for (int nt = 0; nt < 8; ++nt) {
        const float bv = bias2[nt * 16 + lm];
        v8f a;
        #pragma unroll
        for (int v = 0; v < 8; ++v) a[v] = bv;
        acc[nt] = a;
    }

    for (int t = 0; t < TAPS; ++t) {
        if (!flags[t]) continue;                      // data-driven tap skip
        const int dy = t / 3 - 1;
        const int dx = t % 3 - 1;
        const int hs = h0 + dy;                       // block-uniform

        // stage pre-swizzled A_t into LDS (contiguous 64 KB copy)
        __syncthreads();
        {
            const float* At = As + (size_t)t * CIN * COUT;
#if __has_builtin(__builtin_amdgcn_global_load_async_to_lds_b128)
            for (int i = tid * 4; i < CIN * COUT; i += 256 * 4)
                __builtin_amdgcn_global_load_async_to_lds_b128(
                    (AS1 v4i*)(At + i), (AS3 v4i*)&sB[i], 0, 0);
    #if __has_builtin(__builtin_amdgcn_s_wait_asynccnt)
            __builtin_amdgcn_s_wait_asynccnt(0);
    #else
            asm volatile("s_wait_asynccnt 0x0" ::: "memory");
    #endif
#else
            for (int i = tid * 4; i < CIN * COUT; i += 256 * 4)
                *(v4f*)&sB[i] = *(const v4f*)(At + i);
#endif
        }
        __syncthreads();

        if (hs >= 0 && hs < Hn) {                     // block-uniform branch
            const int  wsh = wt + lm + dx;            // per-lane pixel (edges)
            const bool inb = (wsh >= 0) && (wsh < Wn);
            const float* xrow =
                x + (((size_t)b0 * Hn + hs) * Wn + (inb ? wsh : 0)) * CIN + lh * 2;

            #pragma unroll 2
            for (int k0 = 0; k0 < CIN; k0 += 4) {
                // A-frag: lane<16 -> (M=lm, K=k0,k0+1); lane>=16 -> K=k0+2,k0+3
                v2f afrag = (v2f){0.0f, 0.0f};
                if (inb) afrag = *(const v2f*)(xrow + k0);

                const float* brow = &sB[(k0 >> 2) * 512 + lh * 2];
                #pragma unroll
                for (int nt = 0; nt < 8; ++nt) {
                    // B-frag: rows K=k0+2*lh (+1), col n = nt*16+lm -> one b64
                    const v2f bfrag = *(const v2f*)(brow + (nt * 16 + lm) * 4);
                    acc[nt] = __builtin_amdgcn_wmma_f32_16x16x4_f32(
                        false, afrag, false, bfrag,
                        (short)0, acc[nt], false, false);
                }
            }
        }
    }

    // store: tile nt, vgpr v -> element (M = v + 8*lh, N = nt*16 + lm)
    float* orow = out + (size_t)(pixBase + wave * 16) * COUT;
    #pragma unroll
    for (int v = 0; v < 8; ++v) {
        const int m = v + lh * 8;
        #pragma unroll
        for (int nt = 0; nt < 8; ++nt)
            orow[(size_t)m * COUT + nt * 16 + lm] = acc[nt][v];
    }
}

// ---------------------------------------------------------------------------
extern "C" void kernel_launch(void* const* d_in, const int* in_sizes, int n_in,
                              void* d_out, int out_size, void* d_ws, size_t ws_size,
                              hipStream_t stream) {
    const float* x    = (const float*)d_in[0];   // (8,256,256,128)
    const float* kern = (const float*)d_in[1];   // (3,3,128,128)
    const float* bias = (const float*)d_in[2];   // (128)
    const float* Wd   = (const float*)d_in[3];   // (128,128)
    float* out = (float*)d_out;

    float* As    = (float*)d_ws;                 // 9*128*128 floats (swizzled)
    float* bias2 = As + TAPS * CIN * COUT;       // 128 floats
    int*   flags = (int*)(bias2 + COUT);         // 9 ints

    fold_kernel<<<TAPS, 256, 0, stream>>>(kern, bias, Wd, As, bias2, flags);

    const int nPix    = Bn * Hn * Wn;            // 524288
    const int nGroups = nPix / 128;              // 4096
    conv_wmma_kernel<<<nGroups, 256, 0, stream>>>(x, As, bias2, flags, out);
}